// BertClassifier_37907381354985
// MI455X (gfx1250) — compile-verified
//
#include <hip/hip_runtime.h>
#include <math.h>
#include <stdint.h>

#define B_    16
#define L_    128
#define D_    768
#define H1_   770
#define OUT_  40
#define NSPAN 3405          // number of valid (i,j) span pairs
#define NCOLS 1540          // 2*H1 (P | Q concatenated)
#define LDA   772           // padded LDS row stride: 772%64==4 -> conflict-free

typedef float v2f __attribute__((ext_vector_type(2)));
typedef float v8f __attribute__((ext_vector_type(8)));
typedef __attribute__((address_space(3))) float lds_f;

// ---------------------------------------------------------------------------
// Kernel A: PQ = vecs @ [W1i | W1j]
//   vecs rows m = b*128 + l -> hidden[(b*129 + l + 1)*768]; 16-row tiles never
//   cross a batch boundary, so each A tile is one contiguous 48 KB block.
// Block: 128 threads (4 waves). blockIdx.y = M-tile (16 rows).
// Each wave owns TWO 16-col N tiles (16x32) -> A fragment reused across 2 WMMAs.
// A tile staged into LDS with gfx1250 async-to-LDS (ASYNCcnt) path.
// ---------------------------------------------------------------------------
__global__ __launch_bounds__(128) void pq_gemm(const float* __restrict__ hidden,
                                               const float* __restrict__ W1,
                                               float* __restrict__ PQ) {
  __shared__ float As[16 * LDA];
  const int tid  = threadIdx.x;
  const int lane = tid & 31;
  const int wave = tid >> 5;
  const int m_base = blockIdx.y * 16;
  const int n0     = blockIdx.x * 128 + wave * 32;   // two 16-col tiles per wave

  // ---- async-stage A tile (16 x 768 fp32, contiguous) into LDS ----
  {
    const int b = m_base >> 7, l0 = m_base & 127;
    const float* gbase = hidden + (size_t)(b * (L_ + 1) + l0 + 1) * D_;
    lds_f* As3 = (lds_f*)As;
    // 16*192 = 3072 float4 lane-transfers, 24 per thread
    for (int i4 = tid; i4 < 16 * (D_ / 4); i4 += 128) {
      const int r = i4 / (D_ / 4), c4 = i4 - r * (D_ / 4);
      const unsigned ldsAddr = (unsigned)(uintptr_t)(As3 + (r * LDA + c4 * 4));
      const float* g = gbase + (size_t)i4 * 4;
      asm volatile("global_load_async_to_lds_b128 %0, %1, off"
                   :: "v"(ldsAddr), "v"(g) : "memory");
    }
    asm volatile("s_wait_asynccnt 0" ::: "memory");
  }
  __syncthreads();

  // fragment addressing per CDNA5 ISA layouts (wave32)
  const int am   = (lane & 15) * LDA;      // A row base in LDS
  const int koff = (lane >> 4) * 2;        // lanes 16-31 carry K+2 / K+3
  const int n1   = n0 + (lane & 15);
  const int n2   = n1 + 16;
  const bool ok1 = (n1 < NCOLS), ok2 = (n2 < NCOLS);
  // clamp OOB lanes to column 0: loads stay in-bounds, results are discarded
  const int c1 = ok1 ? n1 : 0;
  const int c2 = ok2 ? n2 : 0;
  const float* B1 = (c1 < H1_) ? (W1 + c1) : (W1 + (size_t)D_ * H1_ + (c1 - H1_));
  const float* B2 = (c2 < H1_) ? (W1 + c2) : (W1 + (size_t)D_ * H1_ + (c2 - H1_));

  v8f acc1 = {}, acc2 = {};
  #pragma unroll 4
  for (int k0 = 0; k0 < D_; k0 += 4) {
    const int kA = k0 + koff;
    v2f a, b1, b2;
    a.x  = As[am + kA];
    a.y  = As[am + kA + 1];
    b1.x = B1[(size_t)kA * H1_];
    b1.y = B1[(size_t)(kA + 1) * H1_];
    b2.x = B2[(size_t)kA * H1_];
    b2.y = B2[(size_t)(kA + 1) * H1_];
    acc1 = __builtin_amdgcn_wmma_f32_16x16x4_f32(false, a, false, b1,
                                                 (short)0, acc1, false, false);
    acc2 = __builtin_amdgcn_wmma_f32_16x16x4_f32(false, a, false, b2,
                                                 (short)0, acc2, false, false);
  }

  // D layout: VGPR r, lanes 0-15 -> M=r, lanes 16-31 -> M=r+8; N = lane&15
  const int rowb = m_base + (lane >> 4) * 8;
  if (ok1) {
    #pragma unroll
    for (int r = 0; r < 8; ++r)
      PQ[(size_t)(rowb + r) * NCOLS + n1] = acc1[r];
  }
  if (ok2) {
    #pragma unroll
    for (int r = 0; r < 8; ++r)
      PQ[(size_t)(rowb + r) * NCOLS + n2] = acc2[r];
  }
}

// ---------------------------------------------------------------------------
// Map flat span index n -> (i, j).  Pairs are row-major over the mask:
// i<=98 have 30 pairs each (2970 total); i>=99 have 128-i pairs.
// ---------------------------------------------------------------------------
__device__ __forceinline__ void span_from_n(int n, int& i, int& j) {
  if (n < 2970) {
    i = n / 30;
    j = i + (n - i * 30);
  } else {
    int r = n - 2970;
    i = 99;
    int c = 29;
    while (r >= c) { r -= c; --c; ++i; }
    j = i + r;
  }
}

// ---------------------------------------------------------------------------
// Kernel B: per block of 16 span-rows:
//   phase 1: h[m,k] = relu(P[b,i,k] + Q[b,j,k] + ind*w1c[k] + b1[k]) -> LDS
//            (k=770,771 zero-padded so the WMMA K-tail needs no guards)
//   phase 2: logits = h @ W2 via fp32 WMMA (waves 0..2, N tiles 0..2, 40->48)
//   phase 3: log_softmax over 40 classes, store output
// ---------------------------------------------------------------------------
__global__ __launch_bounds__(128) void head_kernel(const float* __restrict__ PQ,
                                                   const float* __restrict__ W1,
                                                   const float* __restrict__ b1,
                                                   const float* __restrict__ W2,
                                                   const float* __restrict__ b2,
                                                   const int*   __restrict__ spans,
                                                   float* __restrict__ out) {
  __shared__ float Hs[16 * LDA];   // 49.4 KB
  __shared__ float Ls[16 * 48];    //  3.0 KB logits tile (N padded to 48)
  const int tid  = threadIdx.x;
  const int lane = tid & 31;
  const int wave = tid >> 5;
  const long row_base = (long)blockIdx.x * 16;

  // ---- phase 1: build ReLU'd h tile in LDS ----
  for (int idx = tid; idx < 16 * LDA; idx += 128) {
    const int m = idx / LDA, k = idx - m * LDA;
    const long grow = row_base + m;
    float v = 0.0f;
    if (grow < (long)B_ * NSPAN && k < H1_) {
      const int b  = (int)(grow / NSPAN);
      const int nn = (int)(grow - (long)b * NSPAN);
      int i, j; span_from_n(nn, i, j);
      const int s = spans[b * 2], e = spans[b * 2 + 1];
      const float ind = (i == s && j == e) ? 2.0f
                      : ((i >= s && j <= e) ? 1.0f : 0.0f);
      const float h = PQ[(size_t)(b * L_ + i) * NCOLS + k]
                    + PQ[(size_t)(b * L_ + j) * NCOLS + H1_ + k]
                    + ind * W1[(size_t)(2 * D_) * H1_ + k]
                    + b1[k];
      v = fmaxf(h, 0.0f);
    }
    Hs[idx] = v;
  }
  __syncthreads();

  // ---- phase 2: logits tile via WMMA; waves 0..2 each own one 16-col N tile
  if (wave < 3) {
    const int n    = wave * 16 + (lane & 15);
    const int ncl  = (n < OUT_) ? n : 0;     // clamp: OOB cols never read back
    const int am   = (lane & 15) * LDA;
    const int koff = (lane >> 4) * 2;
    v8f acc = {};
    #pragma unroll 4
    for (int k0 = 0; k0 < LDA; k0 += 4) {
      const int kA = k0 + koff;
      const int kc = (kA <= D_) ? kA : D_;   // k>=770: A==0, value irrelevant
      v2f a, b;
      a.x = Hs[am + kA];
      a.y = Hs[am + kA + 1];
      b.x = W2[(size_t)kc * OUT_ + ncl];
      b.y = W2[(size_t)(kc + 1) * OUT_ + ncl];
      acc = __builtin_amdgcn_wmma_f32_16x16x4_f32(false, a, false, b,
                                                  (short)0, acc, false, false);
    }
    const int rbase = (lane >> 4) * 8;
    const int col   = wave * 16 + (lane & 15);
    #pragma unroll
    for (int r = 0; r < 8; ++r)
      Ls[(rbase + r) * 48 + col] = acc[r];
  }
  __syncthreads();

  // ---- phase 3: log-softmax per row, store ----
  if (tid < 16) {
    const long grow = row_base + tid;
    if (grow < (long)B_ * NSPAN) {
      float mx = -1e30f;
      for (int c = 0; c < OUT_; ++c) mx = fmaxf(mx, Ls[tid * 48 + c] + b2[c]);
      float se = 0.0f;
      for (int c = 0; c < OUT_; ++c) se += __expf(Ls[tid * 48 + c] + b2[c] - mx);
      const float lse = __logf(se) + mx;
      float* o = out + grow * OUT_;
      for (int c = 0; c < OUT_; ++c) o[c] = Ls[tid * 48 + c] + b2[c] - lse;
    }
  }
}

// ---------------------------------------------------------------------------
extern "C" void kernel_launch(void* const* d_in, const int* in_sizes, int n_in,
                              void* d_out, int out_size, void* d_ws, size_t ws_size,
                              hipStream_t stream) {
  (void)in_sizes; (void)n_in; (void)out_size; (void)ws_size;
  const float* hidden = (const float*)d_in[0];   // (16,129,768) f32
  const int*   spans  = (const int*)  d_in[1];   // (16,2) i32
  // d_in[2] token_num (==128) and d_in[3] span mask are compile-time known
  const float* W1     = (const float*)d_in[4];   // (1537,770)
  const float* b1     = (const float*)d_in[5];   // (770,)
  const float* W2     = (const float*)d_in[6];   // (770,40)
  const float* b2     = (const float*)d_in[7];   // (40,)
  float* out = (float*)d_out;                    // (16,3405,40)
  float* PQ  = (float*)d_ws;                     // (2048,1540) = 12.6 MB scratch

  // Kernel A: 13 N-blocks (4 waves x 32 cols) x 128 M-tiles
  dim3 gA((NCOLS + 127) / 128, (B_ * L_) / 16);
  pq_gemm<<<gA, 128, 0, stream>>>(hidden, W1, PQ);

  // Kernel B: one block per 16 span-rows
  const int nblocks = (B_ * NSPAN + 15) / 16;    // 3405
  head_kernel<<<nblocks, 128, 0, stream>>>(PQ, W1, b1, W2, b2, spans, out);
}